// StreamingEMACMVN_50465865728531
// MI455X (gfx1250) — compile-verified
//
#include <hip/hip_runtime.h>
#include <hip/hip_bf16.h>

// Streaming EMA CMVN:
//  K1: per-chunk per-feature S1/S2 via V_WMMA_F32_16X16X4_F32 (A = ones),
//      loads batched per-batch-pair into registers (one wait per 48 loads),
//      4 independent WMMA accumulation chains for hazard/latency cover.
//  K2: tiny sequential EMA scan (256 steps) -> per-chunk mu, rsqrt(var+eps)
//  K3: bandwidth-bound normalize with b128 loads + nontemporal b128 stores

typedef float v2f __attribute__((ext_vector_type(2)));
typedef float v4f __attribute__((ext_vector_type(4)));
typedef float v8f __attribute__((ext_vector_type(8)));

#define B_DIM   64
#define T_DIM   8192
#define F_DIM   80
#define NCHUNK  256      // T / 32
#define CNT     3072.0f  // B_DIM * 48

#define WMMA_F32(A, B, C) \
    __builtin_amdgcn_wmma_f32_16x16x4_f32(false, (A), false, (B), (short)0, (C), false, false)

// ---------------------------------------------------------------------------
// K1: grid (256 chunks, 5 feature tiles), 256 threads = 8 waves.
// Each wave reduces 8 batches x 48 rows x 16 features using WMMA f32 16x16x4.
// ---------------------------------------------------------------------------
__global__ __launch_bounds__(256) void ema_stats_kernel(
    const float* __restrict__ x, float* __restrict__ S1, float* __restrict__ S2)
{
    const int n    = blockIdx.x;          // chunk index
    const int ft   = blockIdx.y;          // feature tile (16 features)
    const int tid  = threadIdx.x;
    const int wave = tid >> 5;
    const int lane = tid & 31;
    const int f    = ft * 16 + (lane & 15);
    const int koff = (lane < 16) ? 0 : 2; // B-matrix: vgpr0 holds K={0,2}, vgpr1 K={1,3}

    v2f ones; ones.x = 1.0f; ones.y = 1.0f;
    v8f c1a = {}, c2a = {};               // chains for even batch
    v8f c1b = {}, c2b = {};               // chains for odd batch

    const int tstart = 32 * n - 16;       // window start (negative only for n==0)

    for (int bp = 0; bp < 4; ++bp) {      // 4 pairs of batches per wave
        const float* __restrict__ xA =
            x + (size_t)(wave * 8 + 2 * bp) * T_DIM * F_DIM;
        const float* __restrict__ xB =
            x + (size_t)(wave * 8 + 2 * bp + 1) * T_DIM * F_DIM;

        float a0[12], a1[12], b0[12], b1[12];

        if (n == 0) {
            // First window: rows [-16,0) are the zero overlap state; feed 0s
            // (zeros contribute exactly 0 to S1 and S2).
            #pragma unroll
            for (int ks = 0; ks < 12; ++ks) {
                if (ks < 4) {
                    a0[ks] = 0.0f; a1[ks] = 0.0f; b0[ks] = 0.0f; b1[ks] = 0.0f;
                } else {
                    const int t0 = tstart + 4 * ks + koff;
                    const float* pa = xA + (size_t)t0 * F_DIM + f;
                    const float* pb = xB + (size_t)t0 * F_DIM + f;
                    a0[ks] = pa[0]; a1[ks] = pa[F_DIM];
                    b0[ks] = pb[0]; b1[ks] = pb[F_DIM];
                }
            }
        } else {
            #pragma unroll
            for (int ks = 0; ks < 12; ++ks) {
                const int t0 = tstart + 4 * ks + koff;
                const float* pa = xA + (size_t)t0 * F_DIM + f;
                const float* pb = xB + (size_t)t0 * F_DIM + f;
                a0[ks] = pa[0]; a1[ks] = pa[F_DIM];
                b0[ks] = pb[0]; b1[ks] = pb[F_DIM];
            }
        }

        // Branch-free WMMA phase: C[m][f] += sum_k B[k][f] with A = ones.
        #pragma unroll
        for (int ks = 0; ks < 12; ++ks) {
            v2f bm, bq;
            bm.x = a0[ks];          bm.y = a1[ks];
            bq.x = a0[ks] * a0[ks]; bq.y = a1[ks] * a1[ks];
            c1a = WMMA_F32(ones, bm, c1a);
            c2a = WMMA_F32(ones, bq, c2a);
            bm.x = b0[ks];          bm.y = b1[ks];
            bq.x = b0[ks] * b0[ks]; bq.y = b1[ks] * b1[ks];
            c1b = WMMA_F32(ones, bm, c1b);
            c2b = WMMA_F32(ones, bq, c2b);
        }
    }

    // Every lane's c[0] holds the column sum for feature ft*16 + (lane&15).
    __shared__ float s1buf[8][16];
    __shared__ float s2buf[8][16];
    if (lane < 16) {
        s1buf[wave][lane] = c1a[0] + c1b[0];
        s2buf[wave][lane] = c2a[0] + c2b[0];
    }
    __syncthreads();
    if (tid < 16) {
        float a1s = 0.0f, a2s = 0.0f;
        #pragma unroll
        for (int w = 0; w < 8; ++w) { a1s += s1buf[w][tid]; a2s += s2buf[w][tid]; }
        S1[n * F_DIM + ft * 16 + tid] = a1s;
        S2[n * F_DIM + ft * 16 + tid] = a2s;
    }
}

// ---------------------------------------------------------------------------
// K2: sequential EMA scan. One block; thread f handles one feature.
// var_n uses the algebraic expansion of mean((stat - mu_{n-1})^2).
// ---------------------------------------------------------------------------
__global__ void ema_scan_kernel(const float* __restrict__ S1, const float* __restrict__ S2,
                                float* __restrict__ MU, float* __restrict__ INV)
{
    const int f = threadIdx.x;
    if (f >= F_DIM) return;
    const float inv_cnt = 1.0f / CNT;
    float mu = 0.0f, var = 1.0f;
    for (int n = 0; n < NCHUNK; ++n) {
        const float s1 = S1[n * F_DIM + f];
        const float s2 = S2[n * F_DIM + f];
        const float cm = s1 * inv_cnt;
        const float cv = s2 * inv_cnt - 2.0f * mu * cm + mu * mu;
        mu  = 0.999f * mu  + 0.001f * cm;
        var = 0.999f * var + 0.001f * cv;
        MU[n * F_DIM + f]  = mu;
        INV[n * F_DIM + f] = rsqrtf(var + 1e-5f);
    }
}

// ---------------------------------------------------------------------------
// K3: streaming normalize. b128 loads (F=80 is 4-aligned); MU/INV are 160KB
// and L2-hot; output stored nontemporally (never re-read). 32-bit indexing.
// ---------------------------------------------------------------------------
__global__ __launch_bounds__(256) void ema_norm_kernel(
    const float* __restrict__ x, const float* __restrict__ MU,
    const float* __restrict__ INV, float* __restrict__ out)
{
    const int total4 = B_DIM * T_DIM * (F_DIM / 4);   // 10,485,760 < 2^31
    const int i = blockIdx.x * blockDim.x + threadIdx.x;
    if (i >= total4) return;
    const int f4  = i % (F_DIM / 4);
    const int row = i / (F_DIM / 4);                  // b*T + t
    const int t   = row & (T_DIM - 1);
    const int n   = t >> 5;

    const v4f xv = ((const v4f*)x)[i];
    const v4f m  = ((const v4f*)MU)[n * (F_DIM / 4) + f4];
    const v4f s  = ((const v4f*)INV)[n * (F_DIM / 4) + f4];
    const v4f o  = (xv - m) * s;
    __builtin_nontemporal_store(o, (v4f*)out + i);
}

extern "C" void kernel_launch(void* const* d_in, const int* in_sizes, int n_in,
                              void* d_out, int out_size, void* d_ws, size_t ws_size,
                              hipStream_t stream) {
    (void)in_sizes; (void)n_in; (void)out_size; (void)ws_size;
    const float* x   = (const float*)d_in[0];
    float*       out = (float*)d_out;

    // Workspace partition: 4 arrays of 256*80 floats = 320 KB total.
    float* S1  = (float*)d_ws;
    float* S2  = S1 + NCHUNK * F_DIM;
    float* MU  = S2 + NCHUNK * F_DIM;
    float* INV = MU + NCHUNK * F_DIM;

    dim3 g1(NCHUNK, 5);
    ema_stats_kernel<<<g1, 256, 0, stream>>>(x, S1, S2);

    ema_scan_kernel<<<1, 128, 0, stream>>>(S1, S2, MU, INV);

    const int total4 = B_DIM * T_DIM * (F_DIM / 4);
    const int blocks = (total4 + 255) / 256;
    ema_norm_kernel<<<blocks, 256, 0, stream>>>(x, MU, INV, out);
}